// RSC_62912680952359
// MI455X (gfx1250) — compile-verified
//
#include <hip/hip_runtime.h>
#include <hip/hip_bf16.h>
#include <stdint.h>

// ---------------------------------------------------------------------------
// RSC feature-dropout kernel for gfx1250 (MI455X).
//   z_grads[b,c,:,:] = W[labels[b],c]/256  (analytic grad; pure row gather)
//   thr[b]           = 1372nd (0-based) order statistic of W[labels[b],:]
//   mult[b,c]        = keep[b] ? 1 : (g > thr ? 0 : 1/0.67)
//   out              = features * mult   (memory-bound streaming pass)
// Roofline: 256 MB HBM traffic / 23.3 TB/s ~= 11 us; kernel A is negligible.
// ---------------------------------------------------------------------------

typedef float        v4f __attribute__((ext_vector_type(4)));
typedef unsigned int v4u __attribute__((ext_vector_type(4)));
typedef int          v4i __attribute__((ext_vector_type(4)));
typedef int          v8i __attribute__((ext_vector_type(8)));

#define C_CH    2048
#define KSEL    1372u          // 0-based order-statistic index (from quantile math)
#define RETAINP 0.67f

// ---------------- JAX threefry2x32 (exact) ----------------
__device__ __forceinline__ void tf2x32(unsigned k0, unsigned k1,
                                       unsigned x0, unsigned x1,
                                       unsigned& o0, unsigned& o1) {
  const unsigned ks2 = k0 ^ k1 ^ 0x1BD11BDAu;
  x0 += k0; x1 += k1;
#define TF_R(r) { x0 += x1; x1 = (x1 << (r)) | (x1 >> (32 - (r))); x1 ^= x0; }
  TF_R(13) TF_R(15) TF_R(26) TF_R(6)   x0 += k1;  x1 += ks2 + 1u;
  TF_R(17) TF_R(29) TF_R(16) TF_R(24)  x0 += ks2; x1 += k0  + 2u;
  TF_R(13) TF_R(15) TF_R(26) TF_R(6)   x0 += k0;  x1 += k1  + 3u;
  TF_R(17) TF_R(29) TF_R(16) TF_R(24)  x0 += k1;  x1 += ks2 + 4u;
  TF_R(13) TF_R(15) TF_R(26) TF_R(6)   x0 += ks2; x1 += k0  + 5u;
#undef TF_R
  o0 = x0; o1 = x1;
}

// jax.random.split(key, 2): counts [0,1,2,3] -> lanes (0,2),(1,3);
// keyA = (o0_lane0, o0_lane1), keyB = (o1_lane0, o1_lane1)
__device__ __forceinline__ void tf_split(unsigned k0, unsigned k1,
                                         unsigned& a0, unsigned& a1,
                                         unsigned& b0, unsigned& b1) {
  unsigned p0, p1, q0, q1;
  tf2x32(k0, k1, 0u, 2u, p0, p1);
  tf2x32(k0, k1, 1u, 3u, q0, q1);
  a0 = p0; a1 = q0;
  b0 = p1; b1 = q1;
}

// _random_bits(key, 32, shape=(64,)): counts iota(64) split halves; lane i does (i, 32+i)
__device__ __forceinline__ unsigned tf_bits64(unsigned k0, unsigned k1, int idx) {
  const unsigned lane = (idx < 32) ? (unsigned)idx : (unsigned)(idx - 32);
  unsigned o0, o1;
  tf2x32(k0, k1, lane, lane + 32u, o0, o1);
  return (idx < 32) ? o0 : o1;
}

// monotonic float -> uint key (ascending key == ascending float)
__device__ __forceinline__ unsigned f2key(float f) {
  unsigned u = __float_as_uint(f);
  return (u & 0x80000000u) ? ~u : (u | 0x80000000u);
}

// ---------------------------------------------------------------------------
// Kernel A: one block per sample. TDM the gathered W row into LDS, radix-select
// the KSEL-th order statistic, derive per-sample threefry masks, emit mult[b,c].
// ---------------------------------------------------------------------------
__global__ void __launch_bounds__(256)
rsc_mask_kernel(const float* __restrict__ W, const int* __restrict__ labels,
                float* __restrict__ mult) {
  __shared__ float    sW[C_CH];        // 8 KB gathered row
  __shared__ unsigned hist[256];
  __shared__ unsigned s_sel[2];        // [0]=key prefix, [1]=remaining rank

  const int b   = blockIdx.x;
  const int tid = threadIdx.x;
  const int label = labels[b];

  // ---- Tensor Data Mover: DMA W[label, 0:2048] (8 KB) global -> LDS ----
  if (tid == 0) {
    const uint64_t ga  = (uint64_t)(uintptr_t)(W + (size_t)label * C_CH);
    const unsigned lds = (unsigned)(uintptr_t)&sW[0];
    // D# group0: count=1 | lds_addr | global_addr[56:0] | type=2 ("image")
    v4u g0 = { 1u,
               lds,
               (unsigned)(ga & 0xFFFFFFFFu),
               (unsigned)((ga >> 32) & 0x01FFFFFFu) | (2u << 30) };
    // D# group1: data_size=2 (4B); tensor_dim0=2048; tensor_dim1=1;
    //            tile_dim0=2048; tile_dim1=1; tensor_dim0_stride=2048
    v8i g1 = { (int)(2u << 16),                 // [17:16] data_size
               (int)((C_CH & 0xFFFFu) << 16),   // [63:48] tensor_dim0 lo16
               (int)(1u << 16),                 // [95:80] tensor_dim1 lo16
               (int)((C_CH & 0xFFFFu) << 16),   // [127:112] tile_dim0
               (int)1,                          // [143:128] tile_dim1
               (int)C_CH,                       // [191:160] dim0_stride lo32
               0, 0 };
    v4i gz4 = { 0, 0, 0, 0 };
    v8i gz8 = { 0, 0, 0, 0, 0, 0, 0, 0 };
    // clang-23 / amdgpu-toolchain 6-arg form:
    //   (g0 v4u, g1 v8i, g2 v4i, g3 v4i, v8i, cpol)
    __builtin_amdgcn_tensor_load_to_lds(g0, g1, gz4, gz4, gz8, 0);
  }
  __builtin_amdgcn_s_wait_tensorcnt(0);
  __syncthreads();

  // ---- radix select: KSEL-th smallest of 2048 keys, 4x 8-bit passes ----
  if (tid == 0) { s_sel[0] = 0u; s_sel[1] = KSEL; }
  __syncthreads();
  for (int pass = 0; pass < 4; ++pass) {
    const int      shift = 24 - 8 * pass;
    const unsigned mhi   = (pass == 0) ? 0u : (0xFFFFFFFFu << (32 - 8 * pass));
    hist[tid] = 0u;
    __syncthreads();
    const unsigned pfx = s_sel[0];
    for (int j = tid; j < C_CH; j += 256) {
      const unsigned key = f2key(sW[j]);
      if ((key & mhi) == pfx) atomicAdd(&hist[(key >> shift) & 0xFFu], 1u);
    }
    __syncthreads();
    if (tid == 0) {
      unsigned kr = s_sel[1], acc = 0u, bin = 0u;
      for (; bin < 255u; ++bin) {
        const unsigned h = hist[bin];
        if (acc + h > kr) break;
        acc += h;
      }
      s_sel[0] = pfx | (bin << shift);
      s_sel[1] = kr - acc;
    }
    __syncthreads();
  }
  const unsigned thr_key = s_sel[0];

  // ---- per-sample keep flag, exact JAX threefry chain from key 42 ----
  bool keep;
  {
    unsigned ku0, ku1, kr0, kr1;  tf_split(0u, 42u, ku0, ku1, kr0, kr1);
    // randint(ku,0,2): k1,k2=split(ku); span=2 => multiplier=0 => lower_bits&1
    unsigned ka0, ka1, kb0, kb1;  tf_split(ku0, ku1, ka0, ka1, kb0, kb1);
    const bool unchanged = (tf_bits64(kb0, kb1, b) & 1u) != 0u;
    // uniform(kr): bitcast(0x3f800000 | bits>>9) - 1
    const unsigned ub = tf_bits64(kr0, kr1, b);
    const float u = __uint_as_float(0x3F800000u | (ub >> 9)) - 1.0f;
    const bool rmask = (u >= RETAINP);
    keep = (!rmask) || unchanged;       // final out == features for this sample
    (void)ka0; (void)ka1;
  }

  // ---- emit per-(b,c) multiplier ----
  const float scale = 1.0f / RETAINP;
  for (int j = tid; j < C_CH; j += 256) {
    const unsigned key = f2key(sW[j]);
    const float v = keep ? 1.0f : ((key > thr_key) ? 0.0f : scale);
    mult[(size_t)b * C_CH + j] = v;
  }
}

// ---------------------------------------------------------------------------
// Kernel B: memory-bound streaming pass. 128-bit nontemporal load/store;
// one float4 per thread; multiplier shared by all 64 float4s of a channel.
// ---------------------------------------------------------------------------
__global__ void __launch_bounds__(256)
rsc_apply_kernel(const float* __restrict__ x, const float* __restrict__ mult,
                 float* __restrict__ y, int n4) {
  const int i = blockIdx.x * 256 + threadIdx.x;
  if (i >= n4) return;
  const float s = mult[i >> 6];                 // (b*2048 + c)
  const v4f* __restrict__ xv = (const v4f*)x;
  v4f v = __builtin_nontemporal_load(&xv[i]);   // TH=NT: streamed once
  v *= s;
  __builtin_nontemporal_store(v, &((v4f*)y)[i]);
}

// ---------------------------------------------------------------------------
extern "C" void kernel_launch(void* const* d_in, const int* in_sizes, int n_in,
                              void* d_out, int out_size, void* d_ws, size_t ws_size,
                              hipStream_t stream) {
  const float* features = (const float*)d_in[0];   // [64,2048,16,16] f32
  const float* W        = (const float*)d_in[1];   // [1000,2048] f32
  const int*   labels   = (const int*)d_in[2];     // [64]
  float*       out      = (float*)d_out;
  float*       mult     = (float*)d_ws;            // 64*2048 f32 = 512 KB scratch

  rsc_mask_kernel<<<64, 256, 0, stream>>>(W, labels, mult);

  const int n4 = out_size / 4;                     // 8,388,608 float4s
  rsc_apply_kernel<<<(n4 + 255) / 256, 256, 0, stream>>>(features, mult, out, n4);
}